// Decoder_12249246728712
// MI455X (gfx1250) — compile-verified
//
#include <hip/hip_runtime.h>

typedef __attribute__((ext_vector_type(2))) float v2f;
typedef __attribute__((ext_vector_type(8))) float v8f;

__device__ __forceinline__ float elu1(float x) {
    return x > 0.0f ? x : (__expf(x) - 1.0f);
}

__device__ __forceinline__ v8f wmma_f32(v2f a, v2f b, v8f c) {
    // V_WMMA_F32_16X16X4_F32 : D = A(16x4) * B(4x16) + C(16x16), fp32
    return __builtin_amdgcn_wmma_f32_16x16x4_f32(false, a, false, b, (short)0, c,
                                                 false, false);
}

__device__ __forceinline__ v2f lds2(const float* p) { return *(const v2f*)p; }

// ---------------------------------------------------------------------------
// Mode A: y = act( conv1d(concat_C(x1,x2), w, K=5, pad=2) [+ res] )
// x1:[NB,CIN1,512] x2:[NB,CIN2,512] w:[COUT,CIN1+CIN2,5] y/res:[NB,COUT,512]
// grid = (NB, 512/64, ceil(COUT/16)), block = 128 (4 waves, 16 l-positions each)
// LDS: sW[k][m16][c] (zero-padded M rows), sX[l68][c] (zero-filled halo)
// ---------------------------------------------------------------------------
template <int CIN1, int CIN2, int COUT, bool DO_ELU, bool DO_RES>
__global__ __launch_bounds__(128) void conv_wmma_A(const float* __restrict__ x1,
                                                   const float* __restrict__ x2,
                                                   const float* __restrict__ res,
                                                   const float* __restrict__ w,
                                                   float* __restrict__ y) {
    constexpr int CT = CIN1 + CIN2;
    __shared__ float sW[5 * 16 * CT];
    __shared__ float sX[68 * CT];

    const int nb    = blockIdx.x;
    const int lbase = blockIdx.y * 64;
    const int O0    = blockIdx.z * 16;
    const int tid   = threadIdx.x;

    // ---- stage weights (transposed to [k][m][c], pad rows zeroed) ----
    for (int idx = tid; idx < 5 * 16 * CT; idx += 128) {
        const int k  = idx / (16 * CT);
        const int rm = idx - k * 16 * CT;
        const int m  = rm / CT;
        const int c  = rm - m * CT;
        const int om = O0 + m;
        const int oc = (om < COUT) ? om : (COUT - 1);          // clamped addr
        const float v = w[oc * (CT * 5) + c * 5 + k];          // unconditional
        sW[idx] = (om < COUT) ? v : 0.0f;                      // v_cndmask
    }
    // ---- stage input tile (transposed to [l][c], halo zeroed) ----
    const float* xb1 = x1 + (size_t)nb * CIN1 * 512;
    for (int idx = tid; idx < 68 * CIN1; idx += 128) {
        const int c  = idx / 68;
        const int li = idx - c * 68;
        const int lx = lbase - 2 + li;
        const int lc = lx < 0 ? 0 : (lx > 511 ? 511 : lx);     // clamped addr
        const float v = xb1[c * 512 + lc];                     // unconditional
        sX[li * CT + c] = (lx == lc) ? v : 0.0f;
    }
    if constexpr (CIN2 > 0) {
        const float* xb2 = x2 + (size_t)nb * CIN2 * 512;
        for (int idx = tid; idx < 68 * CIN2; idx += 128) {
            const int c  = idx / 68;
            const int li = idx - c * 68;
            const int lx = lbase - 2 + li;
            const int lc = lx < 0 ? 0 : (lx > 511 ? 511 : lx);
            const float v = xb2[c * 512 + lc];
            sX[li * CT + CIN1 + c] = (lx == lc) ? v : 0.0f;
        }
    }
    __syncthreads();

    const int wave = tid >> 5;
    const int lane = tid & 31;
    const int n    = lane & 15;  // N (position) index; also A-matrix row m
    const int hi   = lane >> 4;  // K-half select

    v8f d = {};
#pragma unroll
    for (int k = 0; k < 5; ++k) {
        const float* sXrow = &sX[(wave * 16 + n + k) * CT + 2 * hi];
        const float* sWrow = &sW[(k * 16 + n) * CT + 2 * hi];
#pragma unroll
        for (int c0 = 0; c0 < CT; c0 += 4) {
            d = wmma_f32(lds2(sWrow + c0), lds2(sXrow + c0), d);  // ds_load_b64 x2
        }
    }

    const int l0 = lbase + wave * 16;
#pragma unroll
    for (int r = 0; r < 8; ++r) {
        const int mo = O0 + r + 8 * hi;
        if (mo < COUT) {
            float v = d[r];
            const size_t oi = (size_t)nb * COUT * 512 + (size_t)mo * 512 + l0 + n;
            if constexpr (DO_RES) v += res[oi];
            if constexpr (DO_ELU) v = elu1(v);
            y[oi] = v;
        }
    }
}

// ---------------------------------------------------------------------------
// Mode B (chunk merge): y = elu(conv(x1,w1)) + elu(conv(x2,w2))
// x1,x2:[NB,CIN,512] w1,w2:[COUT,CIN,5] y:[NB,COUT,512]
// ---------------------------------------------------------------------------
template <int CIN, int COUT>
__global__ __launch_bounds__(128) void conv_wmma_B(const float* __restrict__ x1,
                                                   const float* __restrict__ w1,
                                                   const float* __restrict__ x2,
                                                   const float* __restrict__ w2,
                                                   float* __restrict__ y) {
    __shared__ float sW1[5 * 16 * CIN];
    __shared__ float sW2[5 * 16 * CIN];
    __shared__ float sX1[68 * CIN];
    __shared__ float sX2[68 * CIN];

    const int nb    = blockIdx.x;
    const int lbase = blockIdx.y * 64;
    const int O0    = blockIdx.z * 16;
    const int tid   = threadIdx.x;

    for (int idx = tid; idx < 5 * 16 * CIN; idx += 128) {
        const int k  = idx / (16 * CIN);
        const int rm = idx - k * 16 * CIN;
        const int m  = rm / CIN;
        const int c  = rm - m * CIN;
        const int om = O0 + m;
        const int oc = (om < COUT) ? om : (COUT - 1);
        const float v1 = w1[oc * (CIN * 5) + c * 5 + k];
        const float v2 = w2[oc * (CIN * 5) + c * 5 + k];
        sW1[idx] = (om < COUT) ? v1 : 0.0f;
        sW2[idx] = (om < COUT) ? v2 : 0.0f;
    }
    const float* xb1 = x1 + (size_t)nb * CIN * 512;
    const float* xb2 = x2 + (size_t)nb * CIN * 512;
    for (int idx = tid; idx < 68 * CIN; idx += 128) {
        const int c  = idx / 68;
        const int li = idx - c * 68;
        const int lx = lbase - 2 + li;
        const int lc = lx < 0 ? 0 : (lx > 511 ? 511 : lx);
        const float v1 = xb1[c * 512 + lc];
        const float v2 = xb2[c * 512 + lc];
        sX1[li * CIN + c] = (lx == lc) ? v1 : 0.0f;
        sX2[li * CIN + c] = (lx == lc) ? v2 : 0.0f;
    }
    __syncthreads();

    const int wave = tid >> 5;
    const int lane = tid & 31;
    const int n    = lane & 15;
    const int hi   = lane >> 4;

    v8f d1 = {}, d2 = {};
#pragma unroll
    for (int k = 0; k < 5; ++k) {
        const int xo = (wave * 16 + n + k) * CIN + 2 * hi;
        const int wo = (k * 16 + n) * CIN + 2 * hi;
#pragma unroll
        for (int c0 = 0; c0 < CIN; c0 += 4) {
            d1 = wmma_f32(lds2(&sW1[wo + c0]), lds2(&sX1[xo + c0]), d1);
            d2 = wmma_f32(lds2(&sW2[wo + c0]), lds2(&sX2[xo + c0]), d2);
        }
    }

    const int l0 = lbase + wave * 16;
#pragma unroll
    for (int r = 0; r < 8; ++r) {
        const int mo = O0 + r + 8 * hi;
        if (mo < COUT) {
            const size_t oi = (size_t)nb * COUT * 512 + (size_t)mo * 512 + l0 + n;
            y[oi] = elu1(d1[r]) + elu1(d2[r]);
        }
    }
}

// ---------------------------------------------------------------------------
// Projection: circular conv, C=512 over the D axis of out2[B,144,512], L=144
// proj[b,o,p] = sum_{d,k} w_proj[o,d,k] * out2[b,(p+k-2)%144,d]   (o<7)
// grid = (512, 9), block = 32 (single wave, 16-position tile), K chunked by 128
// sW rows m=7..15 alias a zero row -> no guards in the hot loop.
// ---------------------------------------------------------------------------
__global__ __launch_bounds__(32) void proj_wmma(const float* __restrict__ out2,
                                                const float* __restrict__ wp,
                                                float* __restrict__ proj) {
    constexpr int KC = 128;
    __shared__ float sW[5 * 8 * KC];  // [k][m8][c], row 7 = zeros
    __shared__ float sX[20 * KC];     // [pi][c], circular rows

    const int b    = blockIdx.x;
    const int p0   = blockIdx.y * 16;
    const int tid  = threadIdx.x;
    const int n    = tid & 15;
    const int hi   = tid >> 4;
    const int mrow = (n < 7) ? n : 7;  // rows >=7 read the zero row

    const float* xb = out2 + (size_t)b * 144 * 512;

    v8f d = {};
    for (int cc = 0; cc < 512; cc += KC) {
        for (int idx = tid; idx < 5 * 8 * KC; idx += 32) {
            const int k  = idx / (8 * KC);
            const int rm = idx - k * 8 * KC;
            const int m  = rm / KC;
            const int c  = rm - m * KC;
            const int ms = (m < 7) ? m : 0;
            const float v = wp[ms * 2560 + (cc + c) * 5 + k];
            sW[idx] = (m < 7) ? v : 0.0f;
        }
        for (int idx = tid; idx < 20 * KC; idx += 32) {
            const int pi = idx / KC;
            const int c  = idx - pi * KC;
            int p = p0 - 2 + pi;
            p = (p + 144) % 144;                       // circular padding
            sX[idx] = xb[(size_t)p * 512 + cc + c];    // coalesced rows
        }
        __syncthreads();  // single-wave WG: LDS in-order, barrier is ~free

#pragma unroll
        for (int k = 0; k < 5; ++k) {
            const float* sWrow = &sW[(k * 8 + mrow) * KC + 2 * hi];
            const float* sXrow = &sX[(n + k) * KC + 2 * hi];
#pragma unroll
            for (int c0 = 0; c0 < KC; c0 += 4) {
                d = wmma_f32(lds2(sWrow + c0), lds2(sXrow + c0), d);
            }
        }
        __syncthreads();
    }

#pragma unroll
    for (int r = 0; r < 8; ++r) {
        const int mo = r + 8 * hi;
        if (mo < 7) proj[(size_t)b * 1008 + mo * 144 + p0 + n] = d[r];
    }
}

// ---------------------------------------------------------------------------
// Linear over length axis: out[b,p,c] = sum_q proj[b,c,q]*w_lin[p,q] + b_lin[p]
// proj:[B,7,144] w_lin:[144,144] out:[B,144,7]; grid=(512,9), block=32
// A straight from global (144%16==0, no guard); B zero-padded to 16 rows in LDS
// ---------------------------------------------------------------------------
__global__ __launch_bounds__(32) void linear_wmma(const float* __restrict__ proj,
                                                  const float* __restrict__ wl,
                                                  const float* __restrict__ bl,
                                                  float* __restrict__ out) {
    __shared__ float sB[16 * 144];

    const int b   = blockIdx.x;
    const int p0  = blockIdx.y * 16;
    const int tid = threadIdx.x;
    const int n   = tid & 15;
    const int hi  = tid >> 4;

    for (int idx = tid; idx < 16 * 144; idx += 32) {
        const int r = idx / 144;                      // c row
        const int q = idx - r * 144;
        const int rs = (r < 7) ? r : 0;
        const float v = proj[(size_t)b * 1008 + rs * 144 + q];
        sB[idx] = (r < 7) ? v : 0.0f;
    }
    __syncthreads();

    v8f d = {};
    const float* wrow = wl + (size_t)(p0 + n) * 144 + 2 * hi;
    const float* brow = &sB[n * 144 + 2 * hi];
#pragma unroll 4
    for (int k0 = 0; k0 < 144; k0 += 4) {
        const v2f a = *(const v2f*)(wrow + k0);       // global_load_b64
        d = wmma_f32(a, lds2(brow + k0), d);
    }

#pragma unroll
    for (int r = 0; r < 8; ++r) {
        const int p = p0 + r + 8 * hi;                // always < 144
        if (n < 7) out[(size_t)b * 1008 + p * 7 + n] = d[r] + bl[p];
    }
}

// ---------------------------------------------------------------------------
extern "C" void kernel_launch(void* const* d_in, const int* in_sizes, int n_in,
                              void* d_out, int out_size, void* d_ws, size_t ws_size,
                              hipStream_t stream) {
    (void)in_sizes; (void)n_in; (void)out_size; (void)ws_size;
    const float* enc_out_true  = (const float*)d_in[0];
    const float* enc_out_pred  = (const float*)d_in[1];
    const float* lot0          = (const float*)d_in[2];
    const float* lot1          = (const float*)d_in[3];
    const float* lop0          = (const float*)d_in[4];
    const float* lop1          = (const float*)d_in[5];
    const float* w_conv_z1     = (const float*)d_in[6];
    const float* w_conv_output = (const float*)d_in[7];
    const float* w_te0         = (const float*)d_in[8];
    const float* w_te1         = (const float*)d_in[9];
    const float* w_cm0         = (const float*)d_in[10];
    const float* w_cm1         = (const float*)d_in[11];
    const float* w_m0          = (const float*)d_in[12];
    const float* w_m20         = (const float*)d_in[13];
    const float* w_m1          = (const float*)d_in[14];
    const float* w_m21         = (const float*)d_in[15];
    const float* w_proj        = (const float*)d_in[16];
    const float* w_lin         = (const float*)d_in[17];
    const float* b_lin         = (const float*)d_in[18];
    float* out = (float*)d_out;
    float* ws  = (float*)d_ws;

    const size_t szP2 = (size_t)512 * 24 * 512;   // 6 291 456 floats
    const size_t szP1 = (size_t)512 * 48 * 512;   // 12 582 912
    // workspace layout (with reuse): peak ~302 MB
    float* t0   = ws;
    float* out0 = ws + szP2;
    float* lt1  = ws + 2 * szP2;
    float* lt1b = ws + 3 * szP2;
    float* out1 = ws + 4 * szP2;
    float* lt0  = ws;                 // reuse t0+out0   (szP1 == 2*szP2)
    float* lt0b = ws + 2 * szP2;      // reuse lt1+lt1b
    float* out2 = ws + 4 * szP2 + szP1;
    float* proj = ws;                 // reuse (lt0 dead by then)

    const dim3 blk(128);

    // 1) t0 = conv(enc_out_true, w_conv_z1) + enc_out_pred       [512,24,512]
    conv_wmma_A<32, 0, 24, false, true>
        <<<dim3(512, 8, 2), blk, 0, stream>>>(enc_out_true, nullptr, enc_out_pred, w_conv_z1, t0);
    // 2) out0 = elu(conv(t0, w_conv_output))
    conv_wmma_A<24, 0, 24, true, false>
        <<<dim3(512, 8, 2), blk, 0, stream>>>(t0, nullptr, nullptr, w_conv_output, out0);
    // 3) lt1 = elu(conv(lot1, w_te1))
    conv_wmma_A<32, 0, 24, true, false>
        <<<dim3(512, 8, 2), blk, 0, stream>>>(lot1, nullptr, nullptr, w_te1, lt1);
    // 4) lt1b = elu(conv([lt1 ; lop1], w_cm1))
    conv_wmma_A<24, 24, 24, true, false>
        <<<dim3(512, 8, 2), blk, 0, stream>>>(lt1, lop1, nullptr, w_cm1, lt1b);
    // 5) out1 = elu(conv(out0,w_m0)) + elu(conv(lt1b,w_m20))  NB=3072, 4->8
    conv_wmma_B<4, 8>
        <<<dim3(3072, 8, 1), blk, 0, stream>>>(out0, w_m0, lt1b, w_m20, out1);
    // 6) lt0 = elu(conv(lot0, w_te0))                            [512,48,512]
    conv_wmma_A<64, 0, 48, true, false>
        <<<dim3(512, 8, 3), blk, 0, stream>>>(lot0, nullptr, nullptr, w_te0, lt0);
    // 7) lt0b = elu(conv([lt0 ; lop0], w_cm0))
    conv_wmma_A<48, 48, 48, true, false>
        <<<dim3(512, 8, 3), blk, 0, stream>>>(lt0, lop0, nullptr, w_cm0, lt0b);
    // 8) out2 = elu(conv(out1,w_m1)) + elu(conv(lt0b,w_m21)) NB=3072, 8->24
    conv_wmma_B<8, 24>
        <<<dim3(3072, 8, 2), blk, 0, stream>>>(out1, w_m1, lt0b, w_m21, out2);
    // 9) proj = circular conv over D axis                        [512,7,144]
    proj_wmma<<<dim3(512, 9), dim3(32), 0, stream>>>(out2, w_proj, proj);
    // 10) out = proj @ w_lin^T + b_lin                           [512,144,7]
    linear_wmma<<<dim3(512, 9), dim3(32), 0, stream>>>(proj, w_lin, b_lin, out);
}